// Reproj_48988396978542
// MI455X (gfx1250) — compile-verified
//
#include <hip/hip_runtime.h>
#include <hip/hip_bf16.h>

// Bundle-adjustment reprojection residual for MI455X (gfx1250, wave32).
// Bandwidth-bound: ~126 MB compulsory HBM traffic -> ~5.4 us floor @ 23.3 TB/s.
// Camera tables staged into LDS via the CDNA5 Tensor Data Mover (one DMA per
// table, TENSORcnt), with async-to-LDS per-lane fallback; streams use
// non-temporal hints so the 6 MB point array stays L2-resident;
// global_prefetch_b8 runs ahead on the streamed arrays.

typedef float    v2f   __attribute__((ext_vector_type(2)));
typedef unsigned uint4v __attribute__((ext_vector_type(4)));
typedef unsigned uint8v __attribute__((ext_vector_type(8)));

#define PF_DIST 4       // prefetch distance in grid-strides

// ---- TDM descriptor builders (CDNA5 ISA) ----
// Group 0: count=1, lds_addr, global_addr[56:0], type=2 ("image").
static __device__ __forceinline__ uint4v tdm_group0(unsigned lds_addr, const void* gaddr) {
    unsigned long long a = (unsigned long long)gaddr;
    uint4v g0;
    g0[0] = 1u;                                                    // count=1, no gather
    g0[1] = lds_addr;                                              // LDS byte address
    g0[2] = (unsigned)a;                                           // global_addr[31:0]
    g0[3] = (unsigned)((a >> 32) & 0x01FFFFFFull) | 0x80000000u;   // [56:32] | type=2<<30
    return g0;
}
// Group 1: 1-D tile of `nelems` 4-byte elements (nelems must fit in 16 bits).
static __device__ __forceinline__ uint8v tdm_group1(unsigned nelems) {
    uint8v g1;
    g1[0] = 0x00020000u;                                  // wg_mask=0, data_size=2 (4B)
    g1[1] = (nelems & 0xFFFFu) << 16;                     // tensor_dim0[15:0]
    g1[2] = ((nelems >> 16) & 0xFFFFu) | (1u << 16);      // tensor_dim0[31:16] | tensor_dim1=1
    g1[3] = (nelems & 0xFFFFu) << 16;                     // tensor_dim1 hi=0 | tile_dim0
    g1[4] = 1u;                                           // tile_dim1=1, tile_dim2=0
    g1[5] = nelems;                                       // tensor_dim0_stride[31:0]
    g1[6] = 0u;                                           // stride hi / dim1_stride lo
    g1[7] = 0u;
    return g1;
}

__global__ __launch_bounds__(256) void Reproj_48988396978542_kernel(
    const float* __restrict__ observe,  // [n,2]
    const int*   __restrict__ cidx,     // [n]
    const int*   __restrict__ pidx,     // [n]
    const float* __restrict__ K,        // [n_cams,3]  f,k1,k2
    const float* __restrict__ C,        // [n_cams,7]  t(3), q(x,y,z,w)
    const float* __restrict__ P,        // [n_pts,3]
    float*       __restrict__ out,      // [n,2]
    int n, int n_cams)
{
    // LDS layout: [0, n_cams*7) floats = C table, then [n_cams*7, n_cams*10) = K table.
    extern __shared__ float cam[];

    // Workgroup-relative LDS byte offsets (flat LDS aperture keeps offset in addr[31:0]).
    const unsigned ldsC = (unsigned)(unsigned long long)cam;
    const unsigned ldsK = ldsC + (unsigned)n_cams * 28u;

    const int cTot = n_cams * 7;          // floats in C table
    const int kTot = n_cams * 3;          // floats in K table

    if (cTot <= 0xFFFF && kTot <= 0xFFFF) {
        // ---- TDM path: one DMA per table, issued by wave 0 only (TDM ignores EXEC). ----
        if (threadIdx.x < (unsigned)warpSize) {
            uint4v g0c = tdm_group0(ldsC, C);
            uint8v g1c = tdm_group1((unsigned)cTot);
            asm volatile("tensor_load_to_lds %0, %1" :: "s"(g0c), "s"(g1c) : "memory");
            uint4v g0k = tdm_group0(ldsK, K);
            uint8v g1k = tdm_group1((unsigned)kTot);
            asm volatile("tensor_load_to_lds %0, %1" :: "s"(g0k), "s"(g1k) : "memory");
            __builtin_amdgcn_s_wait_tensorcnt(0);
        }
    } else {
        // ---- Fallback: per-lane async-to-LDS copies (ASYNCcnt). ----
        const int c16 = cTot >> 2;
        const int k16 = kTot >> 2;
        for (int e = (int)threadIdx.x; e < c16; e += (int)blockDim.x) {
            unsigned off = (unsigned)e * 16u, la = ldsC + off;
            asm volatile("global_load_async_to_lds_b128 %0, %1, %2"
                         :: "v"(la), "v"(off), "s"(C) : "memory");
        }
        for (int e = (int)threadIdx.x; e < k16; e += (int)blockDim.x) {
            unsigned off = (unsigned)e * 16u, la = ldsK + off;
            asm volatile("global_load_async_to_lds_b128 %0, %1, %2"
                         :: "v"(la), "v"(off), "s"(K) : "memory");
        }
        for (int e = (c16 << 2) + (int)threadIdx.x; e < cTot; e += (int)blockDim.x) {
            unsigned off = (unsigned)e * 4u, la = ldsC + off;
            asm volatile("global_load_async_to_lds_b32 %0, %1, %2"
                         :: "v"(la), "v"(off), "s"(C) : "memory");
        }
        for (int e = (k16 << 2) + (int)threadIdx.x; e < kTot; e += (int)blockDim.x) {
            unsigned off = (unsigned)e * 4u, la = ldsK + off;
            asm volatile("global_load_async_to_lds_b32 %0, %1, %2"
                         :: "v"(la), "v"(off), "s"(K) : "memory");
        }
        asm volatile("s_wait_asynccnt 0x0" ::: "memory");
    }
    __syncthreads();

    const float* __restrict__ camC = cam;                 // 7 floats per camera
    const float* __restrict__ camK = cam + (size_t)cTot;  // 3 floats per camera

    const int stride = (int)(gridDim.x * blockDim.x);
    for (int i = (int)(blockIdx.x * blockDim.x + threadIdx.x); i < n; i += stride) {
        // Prefetch the streamed arrays a few grid-strides ahead (global_prefetch_b8).
        int ipf = i + PF_DIST * stride;
        if (ipf < n) {
            __builtin_prefetch(cidx + ipf, 0, 1);
            __builtin_prefetch(pidx + ipf, 0, 1);
            __builtin_prefetch(observe + 2 * (size_t)ipf, 0, 1);
        }

        // Streamed, used-once data: non-temporal so P stays L2-resident.
        int c = __builtin_nontemporal_load(cidx + i);
        int p = __builtin_nontemporal_load(pidx + i);
        v2f ob = __builtin_nontemporal_load(reinterpret_cast<const v2f*>(observe) + i);

        // Point gather: 6 MB table, L2-resident after first pass (regular temporal).
        const float* Pp = P + 3 * (size_t)p;
        float px = Pp[0], py = Pp[1], pz = Pp[2];

        // Camera gather from LDS.
        const float* cc = camC + c * 7;
        float tx = cc[0], ty = cc[1], tz = cc[2];
        float qx = cc[3], qy = cc[4], qz = cc[5], qw = cc[6];
        const float* ck = camK + c * 3;
        float fo = ck[0], k1 = ck[1], k2 = ck[2];

        // v + 2*(qw*(qv x v) + qv x (qv x v)) + t
        float uvx = qy * pz - qz * py;
        float uvy = qz * px - qx * pz;
        float uvz = qx * py - qy * px;

        float uuvx = qy * uvz - qz * uvy;
        float uuvy = qz * uvx - qx * uvz;
        float uuvz = qx * uvy - qy * uvx;

        float cpx = px + 2.0f * (qw * uvx + uuvx) + tx;
        float cpy = py + 2.0f * (qw * uvy + uuvy) + ty;
        float cpz = pz + 2.0f * (qw * uvz + uuvz) + tz;

        // One precise divide, then multiplies.
        float inv = 1.0f / cpz;
        float nx = -cpx * inv;
        float ny = -cpy * inv;

        float r  = nx * nx + ny * ny;
        float d  = 1.0f + k1 * r + k2 * r * r;
        float fd = fo * d;

        v2f res;
        res.x = fd * nx - ob.x;
        res.y = fd * ny - ob.y;
        __builtin_nontemporal_store(res, reinterpret_cast<v2f*>(out) + i);
    }
}

extern "C" void kernel_launch(void* const* d_in, const int* in_sizes, int n_in,
                              void* d_out, int out_size, void* d_ws, size_t ws_size,
                              hipStream_t stream) {
    const float* observe = (const float*)d_in[0];
    const int*   cidx    = (const int*)d_in[1];
    const int*   pidx    = (const int*)d_in[2];
    const float* K       = (const float*)d_in[3];
    const float* C       = (const float*)d_in[4];
    const float* P       = (const float*)d_in[5];
    float*       out     = (float*)d_out;

    const int n      = in_sizes[1];          // N_OBS
    const int n_cams = in_sizes[3] / 3;      // K is [n_cams, 3]

    // Persistent grid-stride blocks: cap staging cost (~80 KB LDS fill per block).
    int blocks = (n + 256 * 16 - 1) / (256 * 16);
    if (blocks > 1024) blocks = 1024;
    if (blocks < 1) blocks = 1;

    size_t lds_bytes = (size_t)n_cams * 10 * sizeof(float);  // 7 (C) + 3 (K) floats

    Reproj_48988396978542_kernel<<<blocks, 256, lds_bytes, stream>>>(
        observe, cidx, pidx, K, C, P, out, n, n_cams);
}